// HGCN_71296457113849
// MI455X (gfx1250) — compile-verified
//
#include <hip/hip_runtime.h>
#include <math.h>

#define BATCH 512
#define NNODE 256
#define FEAT  64
#define HD    64
#define LS    72   // LDS row stride (floats): 64 + 8 pad to break bank conflicts

typedef __attribute__((ext_vector_type(2))) float v2f;
typedef __attribute__((ext_vector_type(8))) float v8f;

__device__ __forceinline__ v8f wmma_f32(v2f a, v2f b, v8f c) {
  // D = A(16x4) * B(4x16) + C, fp32 — V_WMMA_F32_16X16X4_F32
  return __builtin_amdgcn_wmma_f32_16x16x4_f32(false, a, false, b, (short)0, c,
                                               false, false);
}

__global__ __launch_bounds__(256) void hgcn_fused(
    const float* __restrict__ x, const float* __restrict__ adj,
    const float* __restrict__ mask, const float* __restrict__ Wemb,
    const float* __restrict__ W0, const float* __restrict__ b0,
    const float* __restrict__ W1, const float* __restrict__ b1,
    const float* __restrict__ Wout, const float* __restrict__ bout,
    float* __restrict__ out) {
  __shared__ float sA[NNODE * LS];
  __shared__ float sB[NNODE * LS];

  const int b     = blockIdx.x;
  const int tid   = threadIdx.x;
  const int wave  = tid >> 5;
  const int lane  = tid & 31;
  const int lhalf = lane & 15;
  const int koff  = (lane >> 4) << 1;  // 0 or 2 (K sub-offset for A/B frags)
  const int rbase = (lane >> 4) << 3;  // 0 or 8 (row offset for C/D tiles)

  const float* xg    = x   + (size_t)b * NNODE * FEAT;
  const float* adjb  = adj + (size_t)b * NNODE * NNODE;
  const float* maskb = mask + (size_t)b * NNODE;
  float*       outb  = out + (size_t)b * NNODE * HD;

  // ---- stage x[b] into sA (coalesced) -----------------------------------
  for (int i = tid; i < NNODE * FEAT; i += 256)
    sA[(i >> 6) * LS + (i & 63)] = xg[i];
  __syncthreads();

  // ---- GEMM1: sB = sA @ Wemb  (K=64). One A-frag feeds 4 nt tiles. ------
  for (int mi = 0; mi < 2; ++mi) {
    const int mt = wave * 2 + mi;
    const int m = mt * 16 + lhalf;
    v8f c0 = {0.f,0.f,0.f,0.f,0.f,0.f,0.f,0.f};
    v8f c1 = c0, c2 = c0, c3 = c0;
#pragma unroll 4
    for (int k = 0; k < FEAT; k += 4) {
      v2f a = *(const v2f*)&sA[m * LS + k + koff];
      const float* wr0 = Wemb + (k + koff) * HD + lhalf;
      const float* wr1 = wr0 + HD;
      v2f bf0 = {wr0[0],  wr1[0]};
      v2f bf1 = {wr0[16], wr1[16]};
      v2f bf2 = {wr0[32], wr1[32]};
      v2f bf3 = {wr0[48], wr1[48]};
      c0 = wmma_f32(a, bf0, c0);
      c1 = wmma_f32(a, bf1, c1);
      c2 = wmma_f32(a, bf2, c2);
      c3 = wmma_f32(a, bf3, c3);
    }
    const int r0 = mt * 16 + rbase;
#pragma unroll
    for (int v = 0; v < 8; ++v) {
      float* dr = &sB[(r0 + v) * LS + lhalf];
      dr[0]  = c0[v];
      dr[16] = c1[v];
      dr[32] = c2[v];
      dr[48] = c3[v];
    }
  }
  __syncthreads();

  // ---- rowwise: sA = expmap0(proj_tan0(sB)), thread t owns row t --------
  {
    const float* src = &sB[tid * LS];
    float* dst = &sA[tid * LS];
    float ss = 0.f;
    for (int cc = 1; cc < HD; ++cc) { float v = src[cc]; ss += v * v; }
    float r = sqrtf(ss);
    float rs = fmaxf(r, 1e-6f);
    float sc = sinhf(rs) / rs;
    dst[0] = coshf(r);
    for (int cc = 1; cc < HD; ++cc) dst[cc] = sc * src[cc];
  }
  __syncthreads();

  // ---- two HGCN layers ---------------------------------------------------
  for (int layer = 0; layer < 2; ++layer) {
    const float* W    = layer ? W1 : W0;
    const float* bias = layer ? b1 : b0;

    // logmap0 in place on sA
    {
      float* row = &sA[tid * LS];
      float x0 = fmaxf(row[0], 1.0f + 1e-7f);
      float d = acoshf(x0);
      float ss = 0.f;
      for (int cc = 1; cc < HD; ++cc) { float v = row[cc]; ss += v * v; }
      float sc = d / fmaxf(sqrtf(ss), 1e-6f);
      row[0] = 0.f;
      for (int cc = 1; cc < HD; ++cc) row[cc] *= sc;
    }
    __syncthreads();

    // GEMM2: sB = sA @ W + bias  (K=64)
    for (int mi = 0; mi < 2; ++mi) {
      const int mt = wave * 2 + mi;
      const int m = mt * 16 + lhalf;
      v8f c0 = {0.f,0.f,0.f,0.f,0.f,0.f,0.f,0.f};
      v8f c1 = c0, c2 = c0, c3 = c0;
#pragma unroll 4
      for (int k = 0; k < HD; k += 4) {
        v2f a = *(const v2f*)&sA[m * LS + k + koff];
        const float* wr0 = W + (k + koff) * HD + lhalf;
        const float* wr1 = wr0 + HD;
        v2f bf0 = {wr0[0],  wr1[0]};
        v2f bf1 = {wr0[16], wr1[16]};
        v2f bf2 = {wr0[32], wr1[32]};
        v2f bf3 = {wr0[48], wr1[48]};
        c0 = wmma_f32(a, bf0, c0);
        c1 = wmma_f32(a, bf1, c1);
        c2 = wmma_f32(a, bf2, c2);
        c3 = wmma_f32(a, bf3, c3);
      }
      const int r0 = mt * 16 + rbase;
      const float bv0 = bias[lhalf];
      const float bv1 = bias[16 + lhalf];
      const float bv2 = bias[32 + lhalf];
      const float bv3 = bias[48 + lhalf];
#pragma unroll
      for (int v = 0; v < 8; ++v) {
        float* dr = &sB[(r0 + v) * LS + lhalf];
        dr[0]  = c0[v] + bv0;
        dr[16] = c1[v] + bv1;
        dr[32] = c2[v] + bv2;
        dr[48] = c3[v] + bv3;
      }
    }
    __syncthreads();

    // Aggregation: sA = adj[b] @ sB (K=256) — dominant HBM traffic.
    // One 8-byte adj load feeds 4 independent WMMAs; each adj element is
    // read from global exactly once per block.
    for (int mi = 0; mi < 2; ++mi) {
      const int mt = wave * 2 + mi;
      const float* arow = adjb + (size_t)(mt * 16 + lhalf) * NNODE + koff;
      v8f c0 = {0.f,0.f,0.f,0.f,0.f,0.f,0.f,0.f};
      v8f c1 = c0, c2 = c0, c3 = c0;
#pragma unroll 4
      for (int k = 0; k < NNODE; k += 4) {
        v2f a = *(const v2f*)(arow + k);
        const float* br0 = &sB[(k + koff) * LS + lhalf];
        const float* br1 = br0 + LS;
        v2f bf0 = {br0[0],  br1[0]};
        v2f bf1 = {br0[16], br1[16]};
        v2f bf2 = {br0[32], br1[32]};
        v2f bf3 = {br0[48], br1[48]};
        c0 = wmma_f32(a, bf0, c0);
        c1 = wmma_f32(a, bf1, c1);
        c2 = wmma_f32(a, bf2, c2);
        c3 = wmma_f32(a, bf3, c3);
      }
      const int r0 = mt * 16 + rbase;
#pragma unroll
      for (int v = 0; v < 8; ++v) {
        float* dr = &sA[(r0 + v) * LS + lhalf];
        dr[0]  = c0[v];
        dr[16] = c1[v];
        dr[32] = c2[v];
        dr[48] = c3[v];
      }
    }
    __syncthreads();

    // rowwise: sA = expmap0(proj_tan0(relu(sA)/NORM_FACTOR)) in place
    {
      float* row = &sA[tid * LS];
      float ss = 0.f;
      for (int cc = 1; cc < HD; ++cc) {
        float v = fmaxf(row[cc], 0.f) * 0.01f;
        ss += v * v;
      }
      float r = sqrtf(ss);
      float rs = fmaxf(r, 1e-6f);
      float sc = sinhf(rs) / rs;
      for (int cc = 1; cc < HD; ++cc)
        row[cc] = sc * (fmaxf(row[cc], 0.f) * 0.01f);
      row[0] = coshf(r);
    }
    __syncthreads();
  }

  // ---- final logmap0 in place --------------------------------------------
  {
    float* row = &sA[tid * LS];
    float x0 = fmaxf(row[0], 1.0f + 1e-7f);
    float d = acoshf(x0);
    float ss = 0.f;
    for (int cc = 1; cc < HD; ++cc) { float v = row[cc]; ss += v * v; }
    float sc = d / fmaxf(sqrtf(ss), 1e-6f);
    row[0] = 0.f;
    for (int cc = 1; cc < HD; ++cc) row[cc] *= sc;
  }
  __syncthreads();

  // ---- GEMM3: out = (sA @ Wout + bout) * node_mask -----------------------
  for (int mi = 0; mi < 2; ++mi) {
    const int mt = wave * 2 + mi;
    const int m = mt * 16 + lhalf;
    v8f c0 = {0.f,0.f,0.f,0.f,0.f,0.f,0.f,0.f};
    v8f c1 = c0, c2 = c0, c3 = c0;
#pragma unroll 4
    for (int k = 0; k < HD; k += 4) {
      v2f a = *(const v2f*)&sA[m * LS + k + koff];
      const float* wr0 = Wout + (k + koff) * HD + lhalf;
      const float* wr1 = wr0 + HD;
      v2f bf0 = {wr0[0],  wr1[0]};
      v2f bf1 = {wr0[16], wr1[16]};
      v2f bf2 = {wr0[32], wr1[32]};
      v2f bf3 = {wr0[48], wr1[48]};
      c0 = wmma_f32(a, bf0, c0);
      c1 = wmma_f32(a, bf1, c1);
      c2 = wmma_f32(a, bf2, c2);
      c3 = wmma_f32(a, bf3, c3);
    }
    const int r0 = mt * 16 + rbase;
    const float bv0 = bout[lhalf];
    const float bv1 = bout[16 + lhalf];
    const float bv2 = bout[32 + lhalf];
    const float bv3 = bout[48 + lhalf];
#pragma unroll
    for (int v = 0; v < 8; ++v) {
      const int rr = r0 + v;
      const float mk = maskb[rr];
      float* dr = outb + rr * HD + lhalf;
      dr[0]  = (c0[v] + bv0) * mk;
      dr[16] = (c1[v] + bv1) * mk;
      dr[32] = (c2[v] + bv2) * mk;
      dr[48] = (c3[v] + bv3) * mk;
    }
  }
}

extern "C" void kernel_launch(void* const* d_in, const int* in_sizes, int n_in,
                              void* d_out, int out_size, void* d_ws,
                              size_t ws_size, hipStream_t stream) {
  const float* x    = (const float*)d_in[0];
  const float* adj  = (const float*)d_in[1];
  const float* mask = (const float*)d_in[2];
  const float* Wemb = (const float*)d_in[3];
  const float* W0   = (const float*)d_in[4];
  const float* b0   = (const float*)d_in[5];
  const float* W1   = (const float*)d_in[6];
  const float* b1   = (const float*)d_in[7];
  const float* Wout = (const float*)d_in[8];
  const float* bout = (const float*)d_in[9];
  float* out = (float*)d_out;

  hgcn_fused<<<dim3(BATCH), dim3(256), 0, stream>>>(
      x, adj, mask, Wemb, W0, b0, W1, b1, Wout, bout, out);
}